// GNN_node_26190710571606
// MI455X (gfx1250) — compile-verified
//
#include <hip/hip_runtime.h>
#include <hip/hip_bf16.h>

// ---------------------------------------------------------------------------
// GIN forward (3 GIN layers + 1 fully-adjacent layer), f32 end-to-end.
// Matmuls use V_WMMA_F32_16X16X4_F32 (CDNA5 wave32 WMMA).
// Z-tile staging uses the Tensor Data Mover (tensor_load_to_lds) with
// LDS padding (row stride 99 dwords -> bank-conflict-free column reads).
// The device build REQUIRES the TDM builtin (compile error otherwise), so a
// successful compile confirms the async-tensor path is in the binary.
// ---------------------------------------------------------------------------

typedef float v2f  __attribute__((ext_vector_type(2)));
typedef float v8f  __attribute__((ext_vector_type(8)));
typedef unsigned int u32x4 __attribute__((ext_vector_type(4)));
typedef int   i32x4 __attribute__((ext_vector_type(4)));
typedef int   i32x8 __attribute__((ext_vector_type(8)));

#define D_EMB 96
#define D_HID 192
#define BN_EPS 1e-5f

// Padded z-tile layout (written by TDM with pad_interval=32dw, pad_amount=1dw):
// element (r, k) lives at dword  99*r + 33*(k>>5) + (k&31)
#define ZIDX(r, k) (99 * (r) + 33 * ((k) >> 5) + ((k) & 31))
#define HID_STRIDE 196   // 196 % 64 == 4 -> conflict-free column reads

// ------------------------- node embedding ----------------------------------
__global__ __launch_bounds__(256)
void embed_kernel(const int* __restrict__ x, const unsigned char* __restrict__ mask,
                  const float* __restrict__ atab, const float* __restrict__ ntab,
                  float* __restrict__ h, int N_) {
    long long t = (long long)blockIdx.x * blockDim.x + threadIdx.x;
    if (t >= (long long)N_ * 24) return;
    int i = (int)(t / 24), c = (int)(t % 24);
    const float* tab = mask[i] ? atab : ntab;
    const float4 v = *(const float4*)(tab + (size_t)x[i] * D_EMB + c * 4);
    *(float4*)(h + (size_t)i * D_EMB + c * 4) = v;
}

// ------------------------- z = (1+eps) * h ---------------------------------
__global__ __launch_bounds__(256)
void scale_init_kernel(const float* __restrict__ h, float* __restrict__ z,
                       const float* __restrict__ eps_ptr, int eps_idx, int N_) {
    long long t = (long long)blockIdx.x * blockDim.x + threadIdx.x;
    if (t >= (long long)N_ * 24) return;
    float s = 1.0f + eps_ptr[eps_idx];
    int i = (int)(t / 24), c = (int)(t % 24);
    float4 v = *(const float4*)(h + (size_t)i * D_EMB + c * 4);
    v.x *= s; v.y *= s; v.z *= s; v.w *= s;
    *(float4*)(z + (size_t)i * D_EMB + c * 4) = v;
}

// -------------- msg = relu(h[src]+e); z[dst] += msg (atomic) ---------------
__global__ __launch_bounds__(256)
void edge_msg_kernel(const float* __restrict__ h, const int* __restrict__ ei,
                     const int* __restrict__ eattr, const unsigned char* __restrict__ mask_e,
                     const float* __restrict__ btab, const float* __restrict__ etab,
                     float* __restrict__ z, int E_) {
    long long t = (long long)blockIdx.x * blockDim.x + threadIdx.x;
    if (t >= (long long)E_ * 24) return;
    int e = (int)(t / 24), c = (int)(t % 24);
    int s = ei[e];
    int d = ei[E_ + e];
    int a = eattr[e];
    const float* tab = mask_e[e] ? btab : etab;
    const float4 hv = *(const float4*)(h + (size_t)s * D_EMB + c * 4);
    const float4 ev = *(const float4*)(tab + (size_t)a * D_EMB + c * 4);
    float m0 = fmaxf(hv.x + ev.x, 0.0f);
    float m1 = fmaxf(hv.y + ev.y, 0.0f);
    float m2 = fmaxf(hv.z + ev.z, 0.0f);
    float m3 = fmaxf(hv.w + ev.w, 0.0f);
    float* zp = z + (size_t)d * D_EMB + c * 4;
    __hip_atomic_fetch_add(zp + 0, m0, __ATOMIC_RELAXED, __HIP_MEMORY_SCOPE_AGENT);
    __hip_atomic_fetch_add(zp + 1, m1, __ATOMIC_RELAXED, __HIP_MEMORY_SCOPE_AGENT);
    __hip_atomic_fetch_add(zp + 2, m2, __ATOMIC_RELAXED, __HIP_MEMORY_SCOPE_AGENT);
    __hip_atomic_fetch_add(zp + 3, m3, __ATOMIC_RELAXED, __HIP_MEMORY_SCOPE_AGENT);
}

// ------------------------- gsum helpers (FA layer) -------------------------
__global__ __launch_bounds__(256)
void zero_kernel(float* __restrict__ p, int n) {
    int t = blockIdx.x * blockDim.x + threadIdx.x;
    if (t < n) p[t] = 0.0f;
}

__global__ __launch_bounds__(256)
void gsum_kernel(const float* __restrict__ h, const int* __restrict__ batch,
                 float* __restrict__ gsum, int N_) {
    long long t = (long long)blockIdx.x * blockDim.x + threadIdx.x;
    if (t >= (long long)N_ * 24) return;
    int i = (int)(t / 24), c = (int)(t % 24);
    int g = batch[i];
    const float4 v = *(const float4*)(h + (size_t)i * D_EMB + c * 4);
    float* gp = gsum + (size_t)g * D_EMB + c * 4;
    __hip_atomic_fetch_add(gp + 0, v.x, __ATOMIC_RELAXED, __HIP_MEMORY_SCOPE_AGENT);
    __hip_atomic_fetch_add(gp + 1, v.y, __ATOMIC_RELAXED, __HIP_MEMORY_SCOPE_AGENT);
    __hip_atomic_fetch_add(gp + 2, v.z, __ATOMIC_RELAXED, __HIP_MEMORY_SCOPE_AGENT);
    __hip_atomic_fetch_add(gp + 3, v.w, __ATOMIC_RELAXED, __HIP_MEMORY_SCOPE_AGENT);
}

__global__ __launch_bounds__(256)
void fa_z_kernel(const float* __restrict__ h, const float* __restrict__ gsum,
                 const int* __restrict__ batch, const float* __restrict__ fa_eps,
                 float* __restrict__ z, int N_) {
    long long t = (long long)blockIdx.x * blockDim.x + threadIdx.x;
    if (t >= (long long)N_ * 24) return;
    float s = 1.0f + fa_eps[0];
    int i = (int)(t / 24), c = (int)(t % 24);
    const float4 hv = *(const float4*)(h + (size_t)i * D_EMB + c * 4);
    const float4 gv = *(const float4*)(gsum + (size_t)batch[i] * D_EMB + c * 4);
    float4 o;
    o.x = s * hv.x + gv.x; o.y = s * hv.y + gv.y;
    o.z = s * hv.z + gv.z; o.w = s * hv.w + gv.w;
    *(float4*)(z + (size_t)i * D_EMB + c * 4) = o;
}

// ---------------------------------------------------------------------------
// MLP: hout = BN(relu(BN(z@W1+b1)) @ W2 + b2) [optional final relu]
// One wave per 16-row tile, 4 waves per block. WMMA f32 16x16x4.
//   A frag (16x4): lanes 0..15 row=lane, VGPR0=K, VGPR1=K+1; lanes 16..31 K+2,K+3
//   B frag (4x16): col = lane&15, K rows mirrored like A
//   C frag (16x16): c[j] = D[j + 8*(lane>>4)][lane&15]
// Z-tile is DMA'd into LDS by the Tensor Data Mover (per-wave descriptor),
// with 1-dword padding every 32 dwords -> row stride 99 (conflict-free).
// ---------------------------------------------------------------------------
__global__ __launch_bounds__(128)
void mlp_wmma_kernel(const float* __restrict__ zin, float* __restrict__ hout,
                     const float* __restrict__ W1, const float* __restrict__ b1,
                     const float* __restrict__ g1, const float* __restrict__ be1,
                     const float* __restrict__ m1, const float* __restrict__ v1,
                     const float* __restrict__ W2, const float* __restrict__ b2,
                     const float* __restrict__ og, const float* __restrict__ ob,
                     const float* __restrict__ om, const float* __restrict__ ov,
                     int n_rows, int final_relu) {
    __shared__ float lds_buf[4][3136];   // 50 KB: z-tile (<=1584 dw) / hidden (16x196)
    const int wave = threadIdx.x >> 5;
    const int lane = threadIdx.x & 31;
    float* wbase = lds_buf[wave];
    const int r0 = (blockIdx.x * 4 + wave) * 16;

    const int mrow  = lane & 15;        // A row; B/C column
    const int khalf = (lane >> 4) * 2;  // K sub-offset for A/B frags
    const int chalf = (lane >> 4) * 8;  // C row offset

#if defined(__gfx1250__) && __has_builtin(__builtin_amdgcn_tensor_load_to_lds)
    // ---- TDM: DMA this wave's 16x96 z tile into LDS (padded layout) ----
    {
        int rem = n_rows - r0;
        if (rem < 0)  rem = 0;
        if (rem > 16) rem = 16;
        unsigned long long ga =
            (unsigned long long)(uintptr_t)(zin + (size_t)r0 * D_EMB);
        unsigned int lds_addr = (unsigned int)(uintptr_t)(&wbase[0]);

        u32x4 g0;
        g0[0] = 1u;                                   // count=1 (valid user D#)
        g0[1] = lds_addr;                             // LDS byte address
        g0[2] = (unsigned int)(ga & 0xFFFFFFFFu);     // global_addr[31:0]
        g0[3] = (unsigned int)((ga >> 32) & 0x1FFFFFFu) | (2u << 30); // [56:32] | type=2

        i32x8 g1v;
        g1v[0] = (2 << 16)        // data_size = 4 bytes
               | (1 << 20)        // pad_enable
               | (4 << 22)        // pad_interval code 4 = 32 dwords
               | (0 << 25);       // pad_amount  code 0 = 1 dword
        g1v[1] = (int)((96u & 0xFFFFu) << 16);                         // tensor_dim0 lo16
        g1v[2] = (int)(((96u >> 16) & 0xFFFFu) |
                       (((unsigned)rem & 0xFFFFu) << 16));             // dim0 hi | dim1 lo
        g1v[3] = (int)((((unsigned)rem >> 16) & 0xFFFFu) | (96u << 16)); // dim1 hi | tile_dim0=96
        g1v[4] = 16;               // tile_dim1 = 16 rows, tile_dim2 = 0
        g1v[5] = 96;               // tensor_dim0_stride lo32 (row-major, stride 96)
        g1v[6] = 0;
        g1v[7] = 0;

        i32x4 gz = {0, 0, 0, 0};
#if __clang_major__ >= 23
        i32x8 gz8 = {0, 0, 0, 0, 0, 0, 0, 0};
        __builtin_amdgcn_tensor_load_to_lds(g0, g1v, gz, gz, gz8, 0);
#else
        __builtin_amdgcn_tensor_load_to_lds(g0, g1v, gz, gz, 0);
#endif
        __builtin_amdgcn_s_wait_tensorcnt(0);   // per-wave: our DMA is done
    }
#elif !defined(__HIP_DEVICE_COMPILE__)
    // Host-pass parse-only fallback (never executed; host never runs this).
    for (int i = lane; i < 16 * 24; i += 32) {
        int r = i / 24, c4 = i % 24;
        float4 v = make_float4(0.f, 0.f, 0.f, 0.f);
        if (r0 + r < n_rows) v = *(const float4*)(zin + (size_t)(r0 + r) * D_EMB + c4 * 4);
        float* p = wbase + ZIDX(r, c4 * 4);
        p[0] = v.x; p[1] = v.y; p[2] = v.z; p[3] = v.w;
    }
    __syncthreads();
#else
#error "gfx1250 device build requires __builtin_amdgcn_tensor_load_to_lds (TDM path)"
#endif

    // ---- GEMM1: [16x96] @ [96x192] -> 12 C-frags ----
    v8f acc1[12] = {};
    for (int k0 = 0; k0 < D_EMB; k0 += 4) {
        const int kk = k0 + khalf;
        const int aidx = ZIDX(mrow, kk);
        v2f a;
        a.x = wbase[aidx];
        a.y = wbase[aidx + 1];   // kk even -> stays inside the 32-dword chunk
#pragma unroll
        for (int t = 0; t < 12; ++t) {
            const int ncol = t * 16 + mrow;
            v2f b;
            b.x = W1[(size_t)kk * D_HID + ncol];
            b.y = W1[(size_t)(kk + 1) * D_HID + ncol];
            acc1[t] = __builtin_amdgcn_wmma_f32_16x16x4_f32(
                false, a, false, b, (short)0, acc1[t], false, false);
        }
    }
    __syncthreads();  // z tile dead; reuse LDS slice for the hidden tile

    // ---- epilogue 1: bias + BN + relu, write hidden 16x192 (stride 196) ----
#pragma unroll
    for (int t = 0; t < 12; ++t) {
        const int ncol = t * 16 + mrow;
        const float sc  = g1[ncol] * rsqrtf(v1[ncol] + BN_EPS);
        const float off = (b1[ncol] - m1[ncol]) * sc + be1[ncol];
#pragma unroll
        for (int j = 0; j < 8; ++j) {
            float val = fmaxf(acc1[t][j] * sc + off, 0.0f);
            wbase[(j + chalf) * HID_STRIDE + ncol] = val;
        }
    }
    __syncthreads();

    // ---- GEMM2: [16x192] @ [192x96] -> 6 C-frags ----
    v8f acc2[6] = {};
    for (int k0 = 0; k0 < D_HID; k0 += 4) {
        const int kk = k0 + khalf;
        v2f a;
        a.x = wbase[mrow * HID_STRIDE + kk];
        a.y = wbase[mrow * HID_STRIDE + kk + 1];
#pragma unroll
        for (int t = 0; t < 6; ++t) {
            const int ncol = t * 16 + mrow;
            v2f b;
            b.x = W2[(size_t)kk * D_EMB + ncol];
            b.y = W2[(size_t)(kk + 1) * D_EMB + ncol];
            acc2[t] = __builtin_amdgcn_wmma_f32_16x16x4_f32(
                false, a, false, b, (short)0, acc2[t], false, false);
        }
    }

    // ---- epilogue 2: bias + outer BN (+relu), store ----
#pragma unroll
    for (int t = 0; t < 6; ++t) {
        const int ncol = t * 16 + mrow;
        const float sc  = og[ncol] * rsqrtf(ov[ncol] + BN_EPS);
        const float off = (b2[ncol] - om[ncol]) * sc + ob[ncol];
#pragma unroll
        for (int j = 0; j < 8; ++j) {
            const int row = r0 + j + chalf;
            float val = acc2[t][j] * sc + off;
            if (final_relu) val = fmaxf(val, 0.0f);
            if (row < n_rows) hout[(size_t)row * D_EMB + ncol] = val;
        }
    }
}

// ---------------------------------------------------------------------------
static inline int gridFor(long long work, int block) {
    return (int)((work + block - 1) / block);
}

extern "C" void kernel_launch(void* const* d_in, const int* in_sizes, int n_in,
                              void* d_out, int out_size, void* d_ws, size_t ws_size,
                              hipStream_t stream) {
    const int N = in_sizes[0];          // 50000
    const int E = in_sizes[2];          // 800000
    const int G = 256;
    const int V = 128;
    const int NL = 3;                   // GIN layers

    const int*           x        = (const int*)d_in[0];
    const int*           ei       = (const int*)d_in[1];
    const int*           eattr    = (const int*)d_in[2];
    const int*           batch    = (const int*)d_in[3];
    const unsigned char* mask_n   = (const unsigned char*)d_in[4];
    const unsigned char* mask_e   = (const unsigned char*)d_in[5];
    const float*         atab     = (const float*)d_in[6];
    const float*         ntab     = (const float*)d_in[7];
    const float*         bond_t   = (const float*)d_in[8];    // [NL,V,96]
    const float*         edge_t   = (const float*)d_in[9];
    const float*         eps_l    = (const float*)d_in[10];   // [NL]
    const float*         W1       = (const float*)d_in[11];   // [NL,96,192]
    const float*         b1       = (const float*)d_in[12];   // [NL,192]
    const float*         g1       = (const float*)d_in[13];
    const float*         be1      = (const float*)d_in[14];
    const float*         m1       = (const float*)d_in[15];
    const float*         v1       = (const float*)d_in[16];
    const float*         W2       = (const float*)d_in[17];   // [NL,192,96]
    const float*         b2       = (const float*)d_in[18];   // [NL,96]
    const float*         fa_eps   = (const float*)d_in[19];
    const float*         faW1     = (const float*)d_in[20];
    const float*         fab1     = (const float*)d_in[21];
    const float*         fag      = (const float*)d_in[22];
    const float*         fabe     = (const float*)d_in[23];
    const float*         fam      = (const float*)d_in[24];
    const float*         fav      = (const float*)d_in[25];
    const float*         faW2     = (const float*)d_in[26];
    const float*         fab2     = (const float*)d_in[27];
    const float*         og       = (const float*)d_in[28];   // [4,96]
    const float*         ob       = (const float*)d_in[29];
    const float*         om       = (const float*)d_in[30];
    const float*         ov       = (const float*)d_in[31];

    float* hbuf = (float*)d_ws;                    // [N,96]
    float* zbuf = hbuf + (size_t)N * D_EMB;        // [N,96]
    float* gsum = zbuf + (size_t)N * D_EMB;        // [G,96]

    const int BK = 256;
    const long long nwork = (long long)N * 24;
    const long long ework = (long long)E * 24;
    const int mlp_blocks = (N + 63) / 64;

    // node embedding
    embed_kernel<<<gridFor(nwork, BK), BK, 0, stream>>>(x, mask_n, atab, ntab, hbuf, N);

    // 3 GIN layers
    for (int l = 0; l < NL; ++l) {
        scale_init_kernel<<<gridFor(nwork, BK), BK, 0, stream>>>(hbuf, zbuf, eps_l, l, N);
        edge_msg_kernel<<<gridFor(ework, BK), BK, 0, stream>>>(
            hbuf, ei, eattr, mask_e,
            bond_t + (size_t)l * V * D_EMB, edge_t + (size_t)l * V * D_EMB, zbuf, E);
        mlp_wmma_kernel<<<mlp_blocks, 128, 0, stream>>>(
            zbuf, hbuf,
            W1 + (size_t)l * D_EMB * D_HID, b1 + (size_t)l * D_HID,
            g1 + (size_t)l * D_HID, be1 + (size_t)l * D_HID,
            m1 + (size_t)l * D_HID, v1 + (size_t)l * D_HID,
            W2 + (size_t)l * D_HID * D_EMB, b2 + (size_t)l * D_EMB,
            og + (size_t)l * D_EMB, ob + (size_t)l * D_EMB,
            om + (size_t)l * D_EMB, ov + (size_t)l * D_EMB,
            N, /*final_relu=*/1);
    }

    // fully-adjacent layer
    zero_kernel<<<gridFor(G * D_EMB, BK), BK, 0, stream>>>(gsum, G * D_EMB);
    gsum_kernel<<<gridFor(nwork, BK), BK, 0, stream>>>(hbuf, batch, gsum, N);
    fa_z_kernel<<<gridFor(nwork, BK), BK, 0, stream>>>(hbuf, gsum, batch, fa_eps, zbuf, N);
    mlp_wmma_kernel<<<mlp_blocks, 128, 0, stream>>>(
        zbuf, (float*)d_out,
        faW1, fab1, fag, fabe, fam, fav, faW2, fab2,
        og + (size_t)3 * D_EMB, ob + (size_t)3 * D_EMB,
        om + (size_t)3 * D_EMB, ov + (size_t)3 * D_EMB,
        N, /*final_relu=*/0);
}